// GATV2_Encoder_79070347920202
// MI455X (gfx1250) — compile-verified
//
#include <hip/hip_runtime.h>
#include <math.h>

// ---------------- problem constants (match reference) ----------------
#define N_NODE   100000
#define D_IN     256
#define HEADS    4
#define D_HEAD   64
#define HC       (HEADS * D_HEAD)   // 256
#define N_EDGE   1600000
#define E_TOT    (N_EDGE + N_NODE)  // 1700000 (self loops appended)
#define NEG_SLOPE 0.2f

typedef __attribute__((ext_vector_type(2))) float v2f;
typedef __attribute__((ext_vector_type(4))) float v4f;
typedef __attribute__((ext_vector_type(8))) float v8f;

// -------------------------------------------------------------------
// helpers
// -------------------------------------------------------------------
__device__ __forceinline__ void edge_sd(const int* __restrict__ ei, int e,
                                        int& s, int& d) {
    if (e < N_EDGE) { s = ei[e]; d = ei[N_EDGE + e]; }
    else            { s = d = e - N_EDGE; }   // self loop
}

// float atomic max via sign-split integer atomics (works with -inf init)
__device__ __forceinline__ void atomicMaxF(float* addr, float val) {
    if (val >= 0.0f) atomicMax((int*)addr, __float_as_int(val));
    else             atomicMin((unsigned int*)addr, (unsigned int)__float_as_int(val));
}

// -------------------------------------------------------------------
// 0) init: zero d_out (agg target), nodeMax = -inf, denom = 0
// -------------------------------------------------------------------
__global__ __launch_bounds__(256) void k_init(float* __restrict__ out,
                                              float* __restrict__ nodeMax,
                                              float* __restrict__ denom) {
    int gid = blockIdx.x * blockDim.x + threadIdx.x;
    if (gid < N_NODE * D_HEAD) out[gid] = 0.0f;
    if (gid < N_NODE * HEADS) { nodeMax[gid] = -INFINITY; denom[gid] = 0.0f; }
}

// -------------------------------------------------------------------
// 1) xl = emb @ W_l, xr = emb @ W_r  via V_WMMA_F32_16X16X4_F32
//    one wave per 16x16 output tile; 8 waves / 256-thread block
// -------------------------------------------------------------------
__global__ __launch_bounds__(256) void k_gemm(const float* __restrict__ emb,
                                              const float* __restrict__ Wl,
                                              const float* __restrict__ Wr,
                                              float* __restrict__ xl,
                                              float* __restrict__ xr) {
    const int TILES_M = N_NODE / 16;   // 6250
    const int TILES_N = HC / 16;       // 16
    const int TOTAL   = TILES_M * TILES_N * 2;

    int waveId = threadIdx.x >> 5;
    int lane   = threadIdx.x & 31;
    int tile   = blockIdx.x * 8 + waveId;      // wave-uniform
    if (tile >= TOTAL) return;                 // EXEC stays all-ones per wave

    int mat = tile / (TILES_M * TILES_N);
    int rem = tile % (TILES_M * TILES_N);
    int tm  = rem / TILES_N;
    int tn  = rem % TILES_N;

    const float* __restrict__ W   = mat ? Wr : Wl;
    float*       __restrict__ OUT = mat ? xr : xl;

    int idx16   = lane & 15;            // A row index / B col index
    int khalf   = (lane >> 4) << 1;     // 0 (lanes 0-15) or 2 (lanes 16-31)
    size_t arow = (size_t)(tm * 16 + idx16) * D_IN;
    int    col  = tn * 16 + idx16;

    v8f c = {};
    #pragma unroll 4
    for (int k = 0; k < D_IN; k += 4) {
        // A frag: lane holds A[m][k+khalf], A[m][k+khalf+1]  (8B-aligned float2)
        v2f a = *(const v2f*)(emb + arow + k + khalf);
        // B frag: lane holds B[k+khalf][col], B[k+khalf+1][col]
        v2f b;
        b[0] = W[(size_t)(k + khalf)     * HC + col];
        b[1] = W[(size_t)(k + khalf + 1) * HC + col];
        c = __builtin_amdgcn_wmma_f32_16x16x4_f32(
                /*neg_a=*/false, a, /*neg_b=*/false, b,
                /*c_mod=*/(short)0, c, /*reuse_a=*/false, /*reuse_b=*/false);
    }

    // C layout: VGPR r -> M = r (lanes 0-15) or 8+r (lanes 16-31), N = lane&15
    int mbase = tm * 16 + ((lane >> 4) << 3);
    #pragma unroll
    for (int r = 0; r < 8; ++r)
        OUT[(size_t)(mbase + r) * HC + col] = c[r];
}

// -------------------------------------------------------------------
// 2) per-edge scores: score[e][h] = att[h] . leakyrelu(xl[src]+xr[dst])
//    one wave per edge; lane owns 8 contiguous channels (head = lane>>3)
// -------------------------------------------------------------------
__global__ __launch_bounds__(256) void k_score(const float* __restrict__ xl,
                                               const float* __restrict__ xr,
                                               const int*   __restrict__ ei,
                                               const float* __restrict__ att,
                                               float* __restrict__ score,
                                               float* __restrict__ nodeMax) {
    int e    = (blockIdx.x * blockDim.x + threadIdx.x) >> 5;   // wave id
    int lane = threadIdx.x & 31;
    if (e >= E_TOT) return;

    int s, d;  edge_sd(ei, e, s, d);

    const v4f* xs = (const v4f*)(xl + (size_t)s * HC + lane * 8);
    const v4f* xd = (const v4f*)(xr + (size_t)d * HC + lane * 8);
    const v4f* at = (const v4f*)(att + lane * 8);
    v4f a0 = xs[0], a1 = xs[1];
    v4f b0 = xd[0], b1 = xd[1];
    v4f t0 = at[0], t1 = at[1];

    float sum = 0.0f;
    #pragma unroll
    for (int j = 0; j < 4; ++j) {
        float v = a0[j] + b0[j];
        v = (v > 0.0f) ? v : NEG_SLOPE * v;
        sum = fmaf(v, t0[j], sum);
        float w = a1[j] + b1[j];
        w = (w > 0.0f) ? w : NEG_SLOPE * w;
        sum = fmaf(w, t1[j], sum);
    }
    // reduce across the 8 lanes of each head
    sum += __shfl_xor(sum, 1, 32);
    sum += __shfl_xor(sum, 2, 32);
    sum += __shfl_xor(sum, 4, 32);

    if ((lane & 7) == 0) {
        int h = lane >> 3;
        score[(size_t)e * HEADS + h] = sum;
        atomicMaxF(&nodeMax[d * HEADS + h], sum);
    }
}

// -------------------------------------------------------------------
// 3) exp(score - max[dst]) in place + denom accumulation
// -------------------------------------------------------------------
__global__ __launch_bounds__(256) void k_exp(const int* __restrict__ ei,
                                             const float* __restrict__ nodeMax,
                                             float* __restrict__ score,
                                             float* __restrict__ denom) {
    int gid = blockIdx.x * blockDim.x + threadIdx.x;
    if (gid >= E_TOT * HEADS) return;
    int e = gid >> 2, h = gid & 3;
    int s, d;  edge_sd(ei, e, s, d);
    float ex = expf(score[gid] - nodeMax[d * HEADS + h]);
    score[gid] = ex;
    atomicAdd(&denom[d * HEADS + h], ex);
}

// -------------------------------------------------------------------
// 4) aggregate: out[dst][c] += (1/H) * alpha[e][h] * xl[src][h][c]
//    wave per edge; cross-head shuffle reduce so lanes 0-7 do the atomics
// -------------------------------------------------------------------
__global__ __launch_bounds__(256) void k_agg(const float* __restrict__ xl,
                                             const int*   __restrict__ ei,
                                             const float* __restrict__ score,
                                             const float* __restrict__ denom,
                                             float* __restrict__ out) {
    int e    = (blockIdx.x * blockDim.x + threadIdx.x) >> 5;
    int lane = threadIdx.x & 31;
    if (e >= E_TOT) return;

    int s, d;  edge_sd(ei, e, s, d);

    int   h     = lane >> 3;
    float ex    = score[(size_t)e * HEADS + h];
    float den   = denom[d * HEADS + h];
    float alpha = 0.25f * ex / (den + 1e-16f);   // 0.25 = mean over heads

    const v4f* xs = (const v4f*)(xl + (size_t)s * HC + lane * 8);
    v4f a0 = xs[0], a1 = xs[1];
    float v[8];
    #pragma unroll
    for (int j = 0; j < 4; ++j) { v[j] = a0[j] * alpha; v[4 + j] = a1[j] * alpha; }

    // lanes L, L^8, L^16, L^24 hold the same output channel c=(L&7)*8+j
    #pragma unroll
    for (int j = 0; j < 8; ++j) {
        v[j] += __shfl_xor(v[j], 8, 32);
        v[j] += __shfl_xor(v[j], 16, 32);
    }
    if (lane < 8) {
        float* dst = out + (size_t)d * D_HEAD + lane * 8;
        #pragma unroll
        for (int j = 0; j < 8; ++j) atomicAdd(dst + j, v[j]);
    }
}

// -------------------------------------------------------------------
// 5) finalize: out = elu(out + bias)
// -------------------------------------------------------------------
__global__ __launch_bounds__(256) void k_final(float* __restrict__ out,
                                               const float* __restrict__ bias) {
    int gid = blockIdx.x * blockDim.x + threadIdx.x;
    if (gid >= N_NODE * D_HEAD) return;
    float v = out[gid] + bias[gid & (D_HEAD - 1)];
    out[gid] = (v > 0.0f) ? v : (expf(v) - 1.0f);
}

// -------------------------------------------------------------------
extern "C" void kernel_launch(void* const* d_in, const int* in_sizes, int n_in,
                              void* d_out, int out_size, void* d_ws, size_t ws_size,
                              hipStream_t stream) {
    // input order: x, edge_index, emb_weight, W_l, W_r, att, bias
    const int*   ei   = (const int*)  d_in[1];
    const float* emb  = (const float*)d_in[2];
    const float* Wl   = (const float*)d_in[3];
    const float* Wr   = (const float*)d_in[4];
    const float* att  = (const float*)d_in[5];
    const float* bias = (const float*)d_in[6];
    float* out = (float*)d_out;

    // workspace carving
    char* ws = (char*)d_ws;
    float* xl      = (float*)(ws);                                   // N*256 f32
    float* xr      = (float*)(ws + (size_t)N_NODE * HC * 4);         // N*256 f32
    float* score   = (float*)(ws + (size_t)N_NODE * HC * 8);         // E_TOT*4 f32
    float* nodeMax = (float*)(ws + (size_t)N_NODE * HC * 8
                                 + (size_t)E_TOT * HEADS * 4);       // N*4 f32
    float* denom   = (float*)((char*)nodeMax + (size_t)N_NODE * HEADS * 4);

    const int B = 256;

    // 0) init out / max / denom
    k_init<<<(N_NODE * D_HEAD + B - 1) / B, B, 0, stream>>>(out, nodeMax, denom);

    // 1) GEMMs (WMMA f32): 6250*16 tiles per matrix, 2 matrices, 8 waves/block
    {
        int totalTiles = (N_NODE / 16) * (HC / 16) * 2;
        int blocks = (totalTiles + 7) / 8;
        k_gemm<<<blocks, B, 0, stream>>>(emb, Wl, Wr, xl, xr);
    }

    // 2) edge scores + segment max (wave per edge)
    k_score<<<(E_TOT + 7) / 8, B, 0, stream>>>(xl, xr, ei, att, score, nodeMax);

    // 3) exp + denom
    k_exp<<<(E_TOT * HEADS + B - 1) / B, B, 0, stream>>>(ei, nodeMax, score, denom);

    // 4) weighted aggregation (wave per edge)
    k_agg<<<(E_TOT + 7) / 8, B, 0, stream>>>(xl, ei, score, denom, out);

    // 5) bias + ELU
    k_final<<<(N_NODE * D_HEAD + B - 1) / B, B, 0, stream>>>(out, bias);
}